// ResNetGNNBase_14697378087198
// MI455X (gfx1250) — compile-verified
//
#include <hip/hip_runtime.h>
#include <hip/hip_bf16.h>

// ---------------------------------------------------------------------------
// GCN (ResNet-GNN) for gfx1250: f32 WMMA GEMMs + L2-resident gather/scatter.
// N=100000 (divisible by 16), E=1.6M, IN=256, HID=128, OUT=64, fp32 end-to-end.
// Workspace layout (floats): [deg/dinv: N][h: N*128][xw: N*128][y: N*128]
//   = ~154 MB.
// ---------------------------------------------------------------------------

constexpr int IND  = 256;
constexpr int HIDN = 128;
constexpr int OUTD = 64;
constexpr int LAYERS = 3;

typedef float v2f __attribute__((ext_vector_type(2)));
typedef float v8f __attribute__((ext_vector_type(8)));

// ---------------- degree / normalization ----------------------------------

__global__ void k_deg_init(float* __restrict__ deg, int n) {
  int i = blockIdx.x * blockDim.x + threadIdx.x;
  if (i < n) deg[i] = 1.0f;   // self-loop contributes 1 to every node
}

__global__ void k_deg_accum(const int* __restrict__ dst, float* __restrict__ deg, int e) {
  int i = blockIdx.x * blockDim.x + threadIdx.x;
  if (i < e) atomicAdd(&deg[dst[i]], 1.0f);
}

__global__ void k_dinv(float* __restrict__ deg, int n) {
  int i = blockIdx.x * blockDim.x + threadIdx.x;
  if (i < n) deg[i] = rsqrtf(deg[i]);   // deg >= 1 always (self loops)
}

// ---------------- WMMA f32 GEMM --------------------------------------------
// out[r0:r0+16, 0:NC] = A[r0:r0+16, 0:K] @ W[0:K, 0:NC] (+ bias)
// Block = NC*2 threads = NC/16 waves; wave w owns the 16-col tile at 16*w.
// A strip (16 x K, contiguous row-major) is staged in LDS once per block.
// Fragment layout per CDNA5 ISA 7.12.2:
//   A 16x4 f32 : lane<16 -> (M=lane, K=k+{0,1}), lane>=16 -> (M=lane-16, K=k+{2,3})
//   B 4x16 f32 : same K split across lane halves, N = lane&15
//   C/D 16x16  : VGPR v -> M = v + 8*(lane>=16), N = lane&15
template <int K, int NC>
__global__ __launch_bounds__(NC * 2)
void k_gemm(const float* __restrict__ A, const float* __restrict__ W,
            const float* __restrict__ bias, float* __restrict__ out) {
  __shared__ float As[16 * K];
  const int r0 = blockIdx.x * 16;
  const int nthreads = NC * 2;
  // 16-row strip of A is fully contiguous (row-major, stride K)
  for (int i = threadIdx.x; i < 16 * K; i += nthreads)
    As[i] = A[(size_t)r0 * K + i];
  __syncthreads();

  const int lane = threadIdx.x & 31;
  const int half = lane >> 4;     // 0 or 1
  const int m    = lane & 15;
  const int col0 = (threadIdx.x >> 5) * 16;

  v8f acc = {};
#pragma unroll 4
  for (int k = 0; k < K; k += 4) {
    v2f a, b;
    a.x = As[m * K + k + 2 * half + 0];
    a.y = As[m * K + k + 2 * half + 1];
    b.x = W[(size_t)(k + 2 * half + 0) * NC + col0 + m];
    b.y = W[(size_t)(k + 2 * half + 1) * NC + col0 + m];
    acc = __builtin_amdgcn_wmma_f32_16x16x4_f32(
        /*neg_a=*/false, a, /*neg_b=*/false, b,
        /*c_mod=*/(short)0, acc, /*reuse_a=*/false, /*reuse_b=*/false);
  }

  const float bv = bias ? bias[col0 + m] : 0.0f;
#pragma unroll
  for (int v = 0; v < 8; ++v)
    out[(size_t)(r0 + v + 8 * half) * NC + col0 + m] = acc[v] + bv;
}

// ---------------- aggregation ----------------------------------------------
// y[n][c] = conv_b[c] + xw[n][c] * dinv[n]^2   (bias + fused self-loop edge)
__global__ void k_agg_init(const float* __restrict__ xw, const float* __restrict__ dinv,
                           const float* __restrict__ bias, float* __restrict__ y, int total) {
  int i = blockIdx.x * blockDim.x + threadIdx.x;
  if (i < total) {
    int node = i >> 7;          // /128
    int c    = i & (HIDN - 1);
    float di = dinv[node];
    y[i] = bias[c] + xw[i] * (di * di);
  }
}

// One wave per edge: 32 lanes x float4 = full 128-channel row.
// xw (51 MB) is L2-resident on MI455X (192 MB L2); atomics land in L2.
__global__ void k_scatter(const int* __restrict__ src, const int* __restrict__ dst,
                          const float* __restrict__ dinv, const float* __restrict__ xw,
                          float* __restrict__ y, int e) {
  int eid = blockIdx.x * (blockDim.x >> 5) + (threadIdx.x >> 5);
  if (eid >= e) return;
  int lane = threadIdx.x & 31;
  int s = src[eid];
  int d = dst[eid];
  float nrm = dinv[s] * dinv[d];
  float4 v = ((const float4*)(xw + (size_t)s * HIDN))[lane];
  float* yp = y + (size_t)d * HIDN + lane * 4;
  atomicAdd(yp + 0, v.x * nrm);
  atomicAdd(yp + 1, v.y * nrm);
  atomicAdd(yp + 2, v.z * nrm);
  atomicAdd(yp + 3, v.w * nrm);
}

// ---------------- LayerNorm + ReLU + residual (one wave32 per node) --------
__global__ void k_ln_relu_res(float* __restrict__ h, const float* __restrict__ y,
                              const float* __restrict__ g, const float* __restrict__ b,
                              int n) {
  int node = blockIdx.x * (blockDim.x >> 5) + (threadIdx.x >> 5);
  if (node >= n) return;
  int lane = threadIdx.x & 31;
  float4 v = ((const float4*)(y + (size_t)node * HIDN))[lane];

  float s = v.x + v.y + v.z + v.w;
#pragma unroll
  for (int m = 16; m >= 1; m >>= 1) s += __shfl_xor(s, m, 32);
  float mu = s * (1.0f / HIDN);

  float dx = v.x - mu, dy = v.y - mu, dz = v.z - mu, dw = v.w - mu;
  float q = dx * dx + dy * dy + dz * dz + dw * dw;
#pragma unroll
  for (int m = 16; m >= 1; m >>= 1) q += __shfl_xor(q, m, 32);
  float rstd = rsqrtf(q * (1.0f / HIDN) + 1e-5f);

  float4 gg = ((const float4*)g)[lane];
  float4 bb = ((const float4*)b)[lane];
  float4* hp = (float4*)(h + (size_t)node * HIDN) + lane;
  float4 hv = *hp;
  hv.x += fmaxf(dx * rstd * gg.x + bb.x, 0.0f);
  hv.y += fmaxf(dy * rstd * gg.y + bb.y, 0.0f);
  hv.z += fmaxf(dz * rstd * gg.z + bb.z, 0.0f);
  hv.w += fmaxf(dw * rstd * gg.w + bb.w, 0.0f);
  *hp = hv;
}

// ---------------------------------------------------------------------------

extern "C" void kernel_launch(void* const* d_in, const int* in_sizes, int n_in,
                              void* d_out, int out_size, void* d_ws, size_t ws_size,
                              hipStream_t stream) {
  const float* x    = (const float*)d_in[0];
  const int*   ei   = (const int*)d_in[1];
  const float* inW  = (const float*)d_in[2];
  const float* inb  = (const float*)d_in[3];
  const float* cW   = (const float*)d_in[4];
  const float* cb   = (const float*)d_in[5];
  const float* lg   = (const float*)d_in[6];
  const float* lb   = (const float*)d_in[7];
  const float* oW   = (const float*)d_in[8];
  const float* ob   = (const float*)d_in[9];

  const int n = in_sizes[0] / IND;   // 100000 (divisible by 16)
  const int e = in_sizes[1] / 2;     // 1600000
  const int* srcp = ei;
  const int* dstp = ei + e;

  float* deg = (float*)d_ws;              // N   (becomes dinv in-place)
  float* h   = deg + n;                   // N*128
  float* xw  = h + (size_t)n * HIDN;      // N*128
  float* y   = xw + (size_t)n * HIDN;     // N*128
  (void)ws_size; (void)n_in; (void)out_size;

  // symmetric normalization
  k_deg_init<<<(n + 255) / 256, 256, 0, stream>>>(deg, n);
  k_deg_accum<<<(e + 255) / 256, 256, 0, stream>>>(dstp, deg, e);
  k_dinv<<<(n + 255) / 256, 256, 0, stream>>>(deg, n);

  // input projection: h = x @ in_W + in_b
  k_gemm<IND, HIDN><<<n / 16, 256, 0, stream>>>(x, inW, inb, h);

  for (int L = 0; L < LAYERS; ++L) {
    // xw = h @ conv_W[L]   (bias folded into aggregation init)
    k_gemm<HIDN, HIDN><<<n / 16, 256, 0, stream>>>(
        h, cW + (size_t)L * HIDN * HIDN, nullptr, xw);
    // y = conv_b + self-loop term
    k_agg_init<<<(n * HIDN + 255) / 256, 256, 0, stream>>>(
        xw, deg, cb + L * HIDN, y, n * HIDN);
    // scatter-add of normalized messages (8 edges per 256-thread block)
    k_scatter<<<(e + 7) / 8, 256, 0, stream>>>(srcp, dstp, deg, xw, y, e);
    // LayerNorm + ReLU + residual into h (8 nodes per block)
    k_ln_relu_res<<<(n + 7) / 8, 256, 0, stream>>>(
        h, y, lg + L * HIDN, lb + L * HIDN, n);
  }

  // output projection: out = h @ out_W + out_b   (4 waves cover 64 cols)
  k_gemm<HIDN, OUTD><<<n / 16, 128, 0, stream>>>(h, oW, ob, (float*)d_out);
}